// DeepGCN_34668976013395
// MI455X (gfx1250) — compile-verified
//
#include <hip/hip_runtime.h>
#include <hip/hip_bf16.h>

typedef __attribute__((ext_vector_type(2))) float v2f;
typedef __attribute__((ext_vector_type(8))) float v8f;

#define N_NODES 50000     // divisible by 16 -> no row masks in GEMM
#define N_EDGES 800000
#define NFEAT   512
#define NHID    128
#define NCLASS  40

// ---------------------------------------------------------------------------
// Float atomic add lowering to global_atomic_add_f32 (relaxed, device scope)
// ---------------------------------------------------------------------------
__device__ __forceinline__ void atomAddF(float* p, float v) {
    __hip_atomic_fetch_add(p, v, __ATOMIC_RELAXED, __HIP_MEMORY_SCOPE_AGENT);
}

// ---------------------------------------------------------------------------
// Weight transpose + zero-pad: Wt[c][k] = (c < N) ? W[k][c] : 0, c in [0,Npad).
// Makes both GEMM operands contiguous along K so fragment loads become
// global_load_b64 with immediate offsets (no per-iteration 64-bit addr math).
// ---------------------------------------------------------------------------
__global__ __launch_bounds__(256) void transpose_pad(
    const float* __restrict__ W, float* __restrict__ Wt, int K, int N, int Npad)
{
    int gid = blockIdx.x * blockDim.x + threadIdx.x;
    if (gid >= Npad * K) return;
    int c = gid / K;
    int k = gid - c * K;
    Wt[gid] = (c < N) ? W[k * N + c] : 0.0f;
}

// ---------------------------------------------------------------------------
// WMMA f32 GEMM: C[M,Nstore] = A[M,K] @ Wt[Npad,K]^T via V_WMMA_F32_16X16X4_F32.
// Requirements: M % 16 == 0, Npad % 64 == 0, K % 16 == 0, K >= 16. One wave
// owns a 16(M) x 64(N) strip with 4 accumulators, reusing the A fragment 4x.
// Depth-2 software pipeline (triple-buffered fragments): K-step k+8 is
// prefetched while computing on k, so every WMMA's operands landed two steps
// earlier and waits retire load-clauses ahead of the compute group.
//
// Fragment layouts (CDNA5 ISA 7.12.2, wave32):
//   A 16x4 : lane (l&15) = M row; half = l>>4; vgpr j holds K = 2*half + j
//   B 4x16 : lane (l&15) = N col; half = l>>4; vgpr j holds K = 2*half + j
//   C 16x16: lane (l&15) = N col; vgpr i holds M = i + 8*half
// EXEC is all-1s through every WMMA (only whole-wave early exit above).
// ---------------------------------------------------------------------------
__global__ __launch_bounds__(256) void gemm_wmma_f32(
    const float* __restrict__ A, const float* __restrict__ Bt,
    float* __restrict__ C, int M, int K, int Npad, int Nstore)
{
    const int lane = threadIdx.x & 31;
    const int wib  = threadIdx.x >> 5;          // wave in block (8 waves/block)
    const int tilesN = Npad >> 6;
    const int tilesM = M >> 4;
    const int wt = blockIdx.x * 8 + wib;
    if (wt >= tilesM * tilesN) return;          // whole-wave exit

    const int tm = wt / tilesN;
    const int tn = wt - tm * tilesN;
    const int nlo  = lane & 15;
    const int half = lane >> 4;

    // Fixed per-lane bases; the k loop adds a pure induction offset.
    const float* __restrict__ Ap  = A  + (tm * 16 + nlo) * K + half * 2;
    const float* __restrict__ Bp0 = Bt + (tn * 64 +  0 + nlo) * K + half * 2;
    const float* __restrict__ Bp1 = Bt + (tn * 64 + 16 + nlo) * K + half * 2;
    const float* __restrict__ Bp2 = Bt + (tn * 64 + 32 + nlo) * K + half * 2;
    const float* __restrict__ Bp3 = Bt + (tn * 64 + 48 + nlo) * K + half * 2;

    v8f acc0 = {}, acc1 = {}, acc2 = {}, acc3 = {};

    // Prologue: fragments for k = 0 (stage X) and k = 4 (stage Y).
    v2f aX  = *(const v2f*)(Ap);
    v2f b0X = *(const v2f*)(Bp0);
    v2f b1X = *(const v2f*)(Bp1);
    v2f b2X = *(const v2f*)(Bp2);
    v2f b3X = *(const v2f*)(Bp3);
    v2f aY  = *(const v2f*)(Ap  + 4);
    v2f b0Y = *(const v2f*)(Bp0 + 4);
    v2f b1Y = *(const v2f*)(Bp1 + 4);
    v2f b2Y = *(const v2f*)(Bp2 + 4);
    v2f b3Y = *(const v2f*)(Bp3 + 4);

#pragma unroll 2
    for (int k = 0; k < K - 8; k += 4) {
        // Prefetch K-step k+8 while the 4 WMMAs on step k execute.
        v2f aZ  = *(const v2f*)(Ap  + k + 8);
        v2f b0Z = *(const v2f*)(Bp0 + k + 8);
        v2f b1Z = *(const v2f*)(Bp1 + k + 8);
        v2f b2Z = *(const v2f*)(Bp2 + k + 8);
        v2f b3Z = *(const v2f*)(Bp3 + k + 8);
        acc0 = __builtin_amdgcn_wmma_f32_16x16x4_f32(false, aX, false, b0X, (short)0, acc0, false, false);
        acc1 = __builtin_amdgcn_wmma_f32_16x16x4_f32(false, aX, false, b1X, (short)0, acc1, false, false);
        acc2 = __builtin_amdgcn_wmma_f32_16x16x4_f32(false, aX, false, b2X, (short)0, acc2, false, false);
        acc3 = __builtin_amdgcn_wmma_f32_16x16x4_f32(false, aX, false, b3X, (short)0, acc3, false, false);
        // Rotate pipeline stages (free under unroll-2 register renaming).
        aX = aY;  b0X = b0Y; b1X = b1Y; b2X = b2Y; b3X = b3Y;
        aY = aZ;  b0Y = b0Z; b1Y = b1Z; b2Y = b2Z; b3Y = b3Z;
    }
    // Epilogue: steps K-8 and K-4.
    acc0 = __builtin_amdgcn_wmma_f32_16x16x4_f32(false, aX, false, b0X, (short)0, acc0, false, false);
    acc1 = __builtin_amdgcn_wmma_f32_16x16x4_f32(false, aX, false, b1X, (short)0, acc1, false, false);
    acc2 = __builtin_amdgcn_wmma_f32_16x16x4_f32(false, aX, false, b2X, (short)0, acc2, false, false);
    acc3 = __builtin_amdgcn_wmma_f32_16x16x4_f32(false, aX, false, b3X, (short)0, acc3, false, false);
    acc0 = __builtin_amdgcn_wmma_f32_16x16x4_f32(false, aY, false, b0Y, (short)0, acc0, false, false);
    acc1 = __builtin_amdgcn_wmma_f32_16x16x4_f32(false, aY, false, b1Y, (short)0, acc1, false, false);
    acc2 = __builtin_amdgcn_wmma_f32_16x16x4_f32(false, aY, false, b2Y, (short)0, acc2, false, false);
    acc3 = __builtin_amdgcn_wmma_f32_16x16x4_f32(false, aY, false, b3Y, (short)0, acc3, false, false);

    const int rbase = tm * 16 + half * 8;
    v8f accs[4] = {acc0, acc1, acc2, acc3};
#pragma unroll
    for (int j = 0; j < 4; ++j) {
        int col = tn * 64 + j * 16 + nlo;
        if (col < Nstore) {                      // only final GEMM (40<64) masks
#pragma unroll
            for (int i = 0; i < 8; ++i)
                C[(rbase + i) * Nstore + col] = accs[j][i];
        }
    }
}

// ---------------------------------------------------------------------------
// SPMM scatter, 4 features/thread: Out[row[e], f..f+3] += val[e]*H[col[e], f..f+3].
// float4 gather (global_load_b128) + 4 native f32 atomics. H/Out are
// L2-resident (25.6 MB vs 192 MB L2) so the random traffic stays on-chip.
// ---------------------------------------------------------------------------
__global__ __launch_bounds__(256) void spmm_scatter4(
    const int* __restrict__ erow, const int* __restrict__ ecol,
    const float* __restrict__ eval, const float* __restrict__ H,
    float* __restrict__ Out, int Fq /*F/4*/, int F, long long total)
{
    long long gid = (long long)blockIdx.x * blockDim.x + threadIdx.x;
    if (gid >= total) return;
    int e  = (int)(gid / Fq);
    int f  = (int)(gid - (long long)e * Fq) * 4;
    int r = erow[e], c = ecol[e];
    float v = eval[e];
    const float4 h = *(const float4*)(H + (size_t)c * F + f);
    float* o = Out + (size_t)r * F + f;
    atomAddF(o + 0, v * h.x);
    atomAddF(o + 1, v * h.y);
    atomAddF(o + 2, v * h.z);
    atomAddF(o + 3, v * h.w);
}

// ---------------------------------------------------------------------------
// pair_norm pass 1: per-column sums (128 cols).
// ---------------------------------------------------------------------------
__global__ __launch_bounds__(NHID) void col_sum(
    const float* __restrict__ X, float* __restrict__ colsum,
    int Mrows, int rowsPerBlock)
{
    int f  = threadIdx.x;
    int r0 = blockIdx.x * rowsPerBlock;
    int r1 = r0 + rowsPerBlock; if (r1 > Mrows) r1 = Mrows;
    float s = 0.0f;
    for (int r = r0; r < r1; ++r) s += X[(size_t)r * NHID + f];
    atomAddF(&colsum[f], s);
}

// ---------------------------------------------------------------------------
// pair_norm pass 2: total sum of squares of column-centered values.
// ---------------------------------------------------------------------------
__global__ __launch_bounds__(NHID) void sum_sq_centered(
    const float* __restrict__ X, const float* __restrict__ colsum,
    float* __restrict__ sq, int Mrows, int rowsPerBlock)
{
    __shared__ float red[NHID];
    int f  = threadIdx.x;
    float mean = colsum[f] * (1.0f / (float)Mrows);
    int r0 = blockIdx.x * rowsPerBlock;
    int r1 = r0 + rowsPerBlock; if (r1 > Mrows) r1 = Mrows;
    float s = 0.0f;
    for (int r = r0; r < r1; ++r) {
        float d = X[(size_t)r * NHID + f] - mean;
        s += d * d;
    }
    red[f] = s;
    __syncthreads();
    for (int stride = NHID / 2; stride > 0; stride >>= 1) {
        if (f < stride) red[f] += red[f + stride];
        __syncthreads();
    }
    if (f == 0) atomAddF(sq, red[0]);
}

// ---------------------------------------------------------------------------
// pair_norm pass 3 + relu + residual:
//   y = relu((x - colmean) * PN_SCALE / sqrt(1e-6 + sumsq/M)) + resid
// (column bias added before pair_norm cancels under centering -> b0/b1/b2
//  are mathematical no-ops and omitted.)
// ---------------------------------------------------------------------------
__global__ __launch_bounds__(256) void pn_relu_residual(
    const float* __restrict__ Xin, const float* __restrict__ colsum,
    const float* __restrict__ sq, const float* __restrict__ resid,
    float* __restrict__ Y, int Mrows)
{
    long long gid = (long long)blockIdx.x * blockDim.x + threadIdx.x;
    long long total = (long long)Mrows * NHID;
    if (gid >= total) return;
    int f = (int)(gid & (NHID - 1));
    float invM  = 1.0f / (float)Mrows;
    float mean  = colsum[f] * invM;
    float scale = rsqrtf(1e-6f + sq[0] * invM);   // PN_SCALE = 1
    float v = (Xin[gid] - mean) * scale;
    v = v > 0.0f ? v : 0.0f;
    if (resid) v += resid[gid];
    Y[gid] = v;
}

// ---------------------------------------------------------------------------
// Initialize final output with broadcast bias (folds b_out ahead of scatter).
// ---------------------------------------------------------------------------
__global__ __launch_bounds__(256) void init_bias(
    float* __restrict__ Out, const float* __restrict__ bias, long long total, int F)
{
    long long gid = (long long)blockIdx.x * blockDim.x + threadIdx.x;
    if (gid >= total) return;
    int f = (int)(gid % F);
    Out[gid] = bias[f];
}

// ---------------------------------------------------------------------------
extern "C" void kernel_launch(void* const* d_in, const int* in_sizes, int n_in,
                              void* d_out, int out_size, void* d_ws, size_t ws_size,
                              hipStream_t stream) {
    const float* x     = (const float*)d_in[0];
    const int*   erow  = (const int*)  d_in[1];
    const int*   ecol  = (const int*)  d_in[2];
    const float* eval  = (const float*)d_in[3];
    const float* W0    = (const float*)d_in[4];
    const float* W1    = (const float*)d_in[6];
    const float* W2    = (const float*)d_in[8];
    const float* Wout  = (const float*)d_in[10];
    const float* bout  = (const float*)d_in[11];
    float* out = (float*)d_out;

    const size_t nh = (size_t)N_NODES * NHID;   // 6.4M floats per buffer
    float* bufA  = (float*)d_ws;
    float* bufB  = bufA + nh;
    float* bufC  = bufB + nh;
    float* stats = bufC + nh;                   // [0..127]=colsum, [128]=sumsq
    float* wtBuf = stats + 256;                 // transposed/padded weights
                                                // (max 128*512 = 64K floats)

    const int rowsPerBlock = 256;
    const int statBlocks   = (N_NODES + rowsPerBlock - 1) / rowsPerBlock;

    auto run_layer = [&](const float* xin, int K, const float* W,
                         float* hg, float* ha, const float* resid, float* yout) {
        // 0) transpose weights: Wt[NHID][K]
        int tN = NHID * K;
        transpose_pad<<<(tN + 255) / 256, 256, 0, stream>>>(W, wtBuf, K, NHID, NHID);
        // 1) dense GEMM via WMMA (exact tiling, no masks)
        int waves = (N_NODES / 16) * (NHID / 64);
        gemm_wmma_f32<<<(waves + 7) / 8, 256, 0, stream>>>(
            xin, wtBuf, hg, N_NODES, K, NHID, NHID);
        // 2) sparse aggregate
        hipMemsetAsync(ha, 0, nh * sizeof(float), stream);
        long long tot = (long long)N_EDGES * (NHID / 4);
        spmm_scatter4<<<(unsigned)((tot + 255) / 256), 256, 0, stream>>>(
            erow, ecol, eval, hg, ha, NHID / 4, NHID, tot);
        // 3) pair_norm stats
        hipMemsetAsync(stats, 0, (NHID + 1) * sizeof(float), stream);
        col_sum<<<statBlocks, NHID, 0, stream>>>(ha, stats, N_NODES, rowsPerBlock);
        sum_sq_centered<<<statBlocks, NHID, 0, stream>>>(ha, stats, stats + NHID,
                                                         N_NODES, rowsPerBlock);
        // 4) normalize + relu + residual
        long long totn = (long long)N_NODES * NHID;
        pn_relu_residual<<<(unsigned)((totn + 255) / 256), 256, 0, stream>>>(
            ha, stats, stats + NHID, resid, yout, N_NODES);
    };

    // Layer 0: x[50000,512] -> bufC          (no residual)
    run_layer(x,    NFEAT, W0, bufA, bufB, nullptr, bufC);
    // Layer 1: bufC -> bufA                  (residual = bufC)
    run_layer(bufC, NHID,  W1, bufA, bufB, bufC,    bufA);
    // Layer 2: bufA -> bufB                  (residual = bufA)
    run_layer(bufA, NHID,  W2, bufB, bufC, bufA,    bufB);

    // Final: out = spmm(bufB @ Wout) + bout   (Wout padded 40 -> 64 cols)
    {
        int tN = 64 * NHID;
        transpose_pad<<<(tN + 255) / 256, 256, 0, stream>>>(Wout, wtBuf, NHID, NCLASS, 64);
        int waves = (N_NODES / 16) * (64 / 64);
        gemm_wmma_f32<<<(waves + 7) / 8, 256, 0, stream>>>(
            bufB, wtBuf, bufA, N_NODES, NHID, 64, NCLASS);
        long long totOut = (long long)N_NODES * NCLASS;
        init_bias<<<(unsigned)((totOut + 255) / 256), 256, 0, stream>>>(
            out, bout, totOut, NCLASS);
        long long tot = (long long)N_EDGES * (NCLASS / 4);
        spmm_scatter4<<<(unsigned)((tot + 255) / 256), 256, 0, stream>>>(
            erow, ecol, eval, bufA, out, NCLASS / 4, NCLASS, tot);
    }
}